// GIN_26645977105018
// MI455X (gfx1250) — compile-verified
//
#include <hip/hip_runtime.h>

// ---------------------------------------------------------------------------
// GIN forward for MI455X (gfx1250). fp32 end-to-end using V_WMMA_F32_16X16X4_F32
// (exact fp32 matrix path; workload is memory-bound so precision costs nothing).
// ---------------------------------------------------------------------------

typedef float v2f __attribute__((ext_vector_type(2)));
typedef float v8f __attribute__((ext_vector_type(8)));

constexpr int   NNODES  = 50000;   // 50000 = 16 * 3125 -> exact 16-row tiling
constexpr int   NEDGES  = 800000;
constexpr int   NFEAT   = 128;
constexpr int   HID     = 64;
constexpr int   NLAYERS = 3;
constexpr int   NGRAPH  = 512;
constexpr int   NCLS    = 10;
constexpr float BNEPS   = 1e-5f;

// ---------------------------------------------------------------------------
// WMMA GEMM:  C[M,64] = op(A)[M,K] @ W[K,64] + bias   (optionally ReLU)
// op(A) = (1+eps)*A + A2 when GIN_FUSE (fuses the GIN self+aggregate update).
//
// Block = 128 threads = 4 waves; wave w computes the 16x16 tile at columns
// [16w, 16w+16) of 16 rows. Grid = M/16 blocks (exact, no divergence -> EXEC
// is all ones for every v_wmma, as the ISA requires).
//
// Fragment layouts (CDNA5 ISA 7.12.2, 32-bit data, wave32):
//   A 16x4 : lanes 0-15 rows, VGPR0={K+0 | K+2 by lane-half}, VGPR1={K+1 | K+3}
//   B 4x16 : mirrored (VGPR0 = rows K+0 / K+2 striped over N=lane&15)
//   C 16x16: VGPR i -> M = i (lanes 0-15) or i+8 (lanes 16-31), N = lane&15
// ---------------------------------------------------------------------------
template <int K, bool GIN_FUSE, bool RELU>
__global__ __launch_bounds__(128) void wmma_gemm_n64(
    const float* __restrict__ A, const float* __restrict__ A2,
    const float* __restrict__ epsp, const float* __restrict__ W,
    const float* __restrict__ bias, float* __restrict__ C)
{
  const int wave  = threadIdx.x >> 5;
  const int lane  = threadIdx.x & 31;
  const int n0    = wave << 4;
  const int row0  = blockIdx.x << 4;
  const int halfk = (lane >> 4) << 1;   // 0 or 2
  const int l15   = lane & 15;

  float epv = 0.0f;
  if (GIN_FUSE) epv = 1.0f + epsp[0];

  // Bias folded into the accumulator: C[m][n] starts at bias[n].
  const float bv = bias[n0 + l15];
  v8f c;
#pragma unroll
  for (int i = 0; i < 8; ++i) c[i] = bv;

  const int    arow   = row0 + l15;
  const float* arowp  = A + (size_t)arow * K;
  const float* arow2p = GIN_FUSE ? (A2 + (size_t)arow * K) : nullptr;

#pragma unroll
  for (int k0 = 0; k0 < K; k0 += 4) {
    const int ka = k0 + halfk;
    v2f a;
    if (GIN_FUSE) {
      a.x = epv * arowp[ka]     + arow2p[ka];
      a.y = epv * arowp[ka + 1] + arow2p[ka + 1];
    } else {
      a = *(const v2f*)(arowp + ka);    // 8B aligned: ka even, K multiple of 4
    }
    v2f b;
    b.x = W[(size_t)ka * 64 + n0 + l15];
    b.y = W[(size_t)(ka + 1) * 64 + n0 + l15];
    // D = A*B + C, fp32, round-to-nearest-even (matches reference fp32 math)
    c = __builtin_amdgcn_wmma_f32_16x16x4_f32(false, a, false, b, (short)0, c,
                                              false, false);
  }

  const int mbase = row0 + ((lane >> 4) << 3);
#pragma unroll
  for (int i = 0; i < 8; ++i) {
    float v = c[i];
    if (RELU) v = fmaxf(v, 0.0f);
    C[(size_t)(mbase + i) * 64 + n0 + l15] = v;
  }
}

// ---------------------------------------------------------------------------
// agg[dst] += h[src] over all edges. 16 threads per edge, float4 per thread ->
// each group of 16 consecutive threads streams one 256B source row coalesced;
// adds resolve as global_atomic_add_f32 in L2 (both buffers fit in 192MB L2).
// ---------------------------------------------------------------------------
__global__ void scatter_edges(const float* __restrict__ h,
                              const int* __restrict__ src,
                              const int* __restrict__ dst,
                              float* __restrict__ agg)
{
  long t = (long)blockIdx.x * blockDim.x + threadIdx.x;
  if (t >= (long)NEDGES * 16) return;
  const int e  = (int)(t >> 4);
  const int c4 = (int)(t & 15);
  const int s = src[e], d = dst[e];
  const float4 v = *(const float4*)(h + (size_t)s * 64 + c4 * 4);
  float* ap = agg + (size_t)d * 64 + c4 * 4;
  atomicAdd(ap + 0, v.x);
  atomicAdd(ap + 1, v.y);
  atomicAdd(ap + 2, v.z);
  atomicAdd(ap + 3, v.w);
}

// Per-channel sum / sum-of-squares for BatchNorm (stats[0:64]=sum, [64:128]=sq)
__global__ void bn_stats(const float* __restrict__ z, float* __restrict__ stats)
{
  const int c    = threadIdx.x & 63;
  const int sub  = threadIdx.x >> 6;                       // 0..3
  const int step = (gridDim.x * blockDim.x) >> 6;
  float s = 0.0f, s2 = 0.0f;
  for (int r = blockIdx.x * (blockDim.x >> 6) + sub; r < NNODES; r += step) {
    const float v = z[(size_t)r * 64 + c];
    s += v;
    s2 += v * v;
  }
  atomicAdd(&stats[c], s);
  atomicAdd(&stats[64 + c], s2);
}

// h = relu( (z - mean) * rsqrt(var + eps) * gamma + beta )
__global__ void bn_apply(const float* __restrict__ z,
                         const float* __restrict__ stats,
                         const float* __restrict__ gamma,
                         const float* __restrict__ beta,
                         float* __restrict__ h)
{
  long t = (long)blockIdx.x * blockDim.x + threadIdx.x;
  if (t >= (long)NNODES * 64) return;
  const int   c    = (int)(t & 63);
  const float mean = stats[c] * (1.0f / NNODES);
  const float var  = stats[64 + c] * (1.0f / NNODES) - mean * mean;
  const float inv  = rsqrtf(var + BNEPS);
  const float v    = (z[t] - mean) * inv * gamma[c] + beta[c];
  h[t] = fmaxf(v, 0.0f);
}

__global__ void zero_f32(float* __restrict__ p, long n)
{
  long t = (long)blockIdx.x * blockDim.x + threadIdx.x;
  if (t < n) p[t] = 0.0f;
}

// pooled[batch[r]] += h[r]; counts[batch[r]] += 1
__global__ void pool_scatter(const float* __restrict__ h,
                             const int* __restrict__ batch,
                             float* __restrict__ pooled,
                             float* __restrict__ counts)
{
  long t = (long)blockIdx.x * blockDim.x + threadIdx.x;
  if (t >= (long)NNODES * 16) return;
  const int r  = (int)(t >> 4);
  const int c4 = (int)(t & 15);
  const int g  = batch[r];
  const float4 v = *(const float4*)(h + (size_t)r * 64 + c4 * 4);
  float* p = pooled + (size_t)g * 64 + c4 * 4;
  atomicAdd(p + 0, v.x);
  atomicAdd(p + 1, v.y);
  atomicAdd(p + 2, v.z);
  atomicAdd(p + 3, v.w);
  if (c4 == 0) atomicAdd(&counts[g], 1.0f);
}

// out[g,c] = (pooled[g,:] / max(count,1)) @ lin_w[:,c] + lin_b[c]   (tiny)
__global__ void final_lin(const float* __restrict__ pooled,
                          const float* __restrict__ counts,
                          const float* __restrict__ lw,
                          const float* __restrict__ lb,
                          float* __restrict__ out)
{
  int t = blockIdx.x * blockDim.x + threadIdx.x;
  if (t >= NGRAPH * 16) return;
  const int g = t >> 4;
  const int c = t & 15;
  if (c >= NCLS) return;
  const float inv = 1.0f / fmaxf(counts[g], 1.0f);
  float acc = lb[c];
#pragma unroll
  for (int k = 0; k < HID; ++k) acc += pooled[(size_t)g * 64 + k] * inv * lw[k * NCLS + c];
  out[g * NCLS + c] = acc;
}

// ---------------------------------------------------------------------------
extern "C" void kernel_launch(void* const* d_in, const int* in_sizes, int n_in,
                              void* d_out, int out_size, void* d_ws,
                              size_t ws_size, hipStream_t stream)
{
  (void)in_sizes; (void)n_in; (void)out_size; (void)ws_size;

  const float* x       = (const float*)d_in[0];
  const int*   eidx    = (const int*)d_in[1];
  const int*   batch   = (const int*)d_in[2];
  const float* enc_w   = (const float*)d_in[3];
  const float* enc_b   = (const float*)d_in[4];
  const float* eps     = (const float*)d_in[5];
  const float* conv_w1 = (const float*)d_in[6];
  const float* conv_b1 = (const float*)d_in[7];
  const float* conv_w2 = (const float*)d_in[8];
  const float* conv_b2 = (const float*)d_in[9];
  const float* gamma   = (const float*)d_in[10];
  const float* beta    = (const float*)d_in[11];
  const float* lin_w   = (const float*)d_in[12];
  const float* lin_b   = (const float*)d_in[13];
  float* out = (float*)d_out;

  const int* srcp = eidx;            // edge_index[0]
  const int* dstp = eidx + NEDGES;   // edge_index[1]

  // Workspace layout (floats): h | agg | tmp | stats(128) | pooled | counts
  float* ws     = (float*)d_ws;
  float* h      = ws;
  float* agg    = h + (size_t)NNODES * 64;
  float* tmp    = agg + (size_t)NNODES * 64;
  float* stats  = tmp + (size_t)NNODES * 64;
  float* pooled = stats + 128;
  float* counts = pooled + (size_t)NGRAPH * 64;

  const int  gemm_grid = NNODES / 16;  // 3125, exact
  const dim3 gblk(128);

  // Encoder: h = x @ enc_w + enc_b
  wmma_gemm_n64<NFEAT, false, false>
      <<<gemm_grid, gblk, 0, stream>>>(x, nullptr, nullptr, enc_w, enc_b, h);

  const long elemsNH = (long)NNODES * 64;
  for (int l = 0; l < NLAYERS; ++l) {
    zero_f32<<<(int)((elemsNH + 255) / 256), 256, 0, stream>>>(agg, elemsNH);
    scatter_edges<<<(NEDGES * 16 + 255) / 256, 256, 0, stream>>>(h, srcp, dstp, agg);
    // tmp = relu(((1+eps)h + agg) @ w1 + b1)   (GIN update fused into A-load)
    wmma_gemm_n64<HID, true, true><<<gemm_grid, gblk, 0, stream>>>(
        h, agg, eps + l, conv_w1 + (size_t)l * HID * HID, conv_b1 + l * HID, tmp);
    // agg = tmp @ w2 + b2   (agg's data is dead; reuse buffer)
    wmma_gemm_n64<HID, false, false><<<gemm_grid, gblk, 0, stream>>>(
        tmp, nullptr, nullptr, conv_w2 + (size_t)l * HID * HID, conv_b2 + l * HID, agg);
    zero_f32<<<1, 128, 0, stream>>>(stats, 128);
    bn_stats<<<256, 256, 0, stream>>>(agg, stats);
    bn_apply<<<(int)((elemsNH + 255) / 256), 256, 0, stream>>>(
        agg, stats, gamma + l * HID, beta + l * HID, h);
  }

  // Global mean pool + head. pooled and counts are contiguous -> one zero pass.
  zero_f32<<<(NGRAPH * 64 + NGRAPH + 255) / 256, 256, 0, stream>>>(
      pooled, NGRAPH * 64 + NGRAPH);
  pool_scatter<<<(NNODES * 16 + 255) / 256, 256, 0, stream>>>(h, batch, pooled, counts);
  final_lin<<<(NGRAPH * 16 + 255) / 256, 256, 0, stream>>>(pooled, counts, lin_w,
                                                           lin_b, out);
}